// MyModel_61933428409434
// MI455X (gfx1250) — compile-verified
//
#include <hip/hip_runtime.h>
#include <hip/hip_bf16.h>

// out = x @ I_4096  ==  out = x  (bit-exact in f32).
// Bandwidth-bound: 256 MiB HBM traffic -> ~11.5 us floor at 23.3 TB/s.
// We still run the matrix op through the CDNA5 matrix pipe by multiplying
// each 16x16 tile by I_16 via V_WMMA_F32_16X16X4_F32 (x*1 + sum(x*0) is exact),
// staged through LDS for fully coalesced global traffic.

typedef float v2f __attribute__((ext_vector_type(2)));
typedef float v4f __attribute__((ext_vector_type(4)));
typedef float v8f __attribute__((ext_vector_type(8)));

#define ROWS 8192
#define COLS 4096
#define SW   512            // strip width (floats)
#define PITCH 516           // LDS row pitch: 516 % 64 == 4 words -> conflict-free
                            // strided A reads; 516*4 B keeps 16B alignment.

__global__ __launch_bounds__(256) void identity_mm_wmma(const float* __restrict__ x,
                                                        float* __restrict__ out) {
    __shared__ float lds[16 * PITCH];          // 33,024 B

    const int t     = threadIdx.x;
    const int strip = blockIdx.x;              // 0..4095
    const int scol  = (strip & 7) * SW;        // 8 horizontal strips
    const int srow  = (strip >> 3) * 16;       // 512 vertical strips

    const float* src = x   + (size_t)srow * COLS + scol;
    float*       dst = out + (size_t)srow * COLS + scol;

    // ---- Stage 1: global -> LDS, coalesced b128 loads (streaming, NT) ----
#pragma unroll
    for (int i = 0; i < 8; ++i) {
        const int idx = t + 256 * i;           // 0..2047 float4 slots
        const int r   = idx >> 7;              // 16 rows, 128 float4/row
        const int c4  = idx & 127;
        const v4f v = __builtin_nontemporal_load(
            reinterpret_cast<const v4f*>(src + (size_t)r * COLS + c4 * 4));
        *reinterpret_cast<v4f*>(&lds[r * PITCH + c4 * 4]) = v;
    }
    __syncthreads();

    // ---- Stage 2: per-wave WMMA tiles: D = A(16x16 of x) * I16 ----
    const int lane = t & 31;
    const int wave = t >> 5;                   // 8 waves; 4 tiles each (32 tiles/strip)
    const int m    = lane & 15;                // A row (lanes 0-15), also D col
    const int half = lane >> 4;                // A: K 0,1 (half=0) vs K 2,3 (half=1)

#pragma unroll
    for (int j = 0; j < 4; ++j) {
        const int col_t = (wave * 4 + j) * 16; // tile column within strip
        v8f acc = {};
#pragma unroll
        for (int c = 0; c < 4; ++c) {
            // A fragment: 16x4 f32 slice, lane holds 2 consecutive K values.
            const float* ap = &lds[m * PITCH + col_t + 4 * c + 2 * half];
            v2f a; a.x = ap[0]; a.y = ap[1];
            // B fragment: rows 4c..4c+3 of I16 (4x16), same lane striping.
            v2f b;
            b.x = (m == (4 * c + 2 * half + 0)) ? 1.0f : 0.0f;
            b.y = (m == (4 * c + 2 * half + 1)) ? 1.0f : 0.0f;
            acc = __builtin_amdgcn_wmma_f32_16x16x4_f32(
                /*neg_a=*/false, a, /*neg_b=*/false, b,
                /*c_mod=*/(short)0, acc, /*reuse_a=*/false, /*reuse_b=*/false);
        }
        // D layout: VGPR rj -> row rj (lanes 0-15) / row rj+8 (lanes 16-31), col = m.
#pragma unroll
        for (int rj = 0; rj < 8; ++rj) {
            lds[(rj + 8 * half) * PITCH + col_t + m] = acc[rj];
        }
    }
    __syncthreads();

    // ---- Stage 3: LDS -> global, coalesced b128 stores (streaming, NT) ----
#pragma unroll
    for (int i = 0; i < 8; ++i) {
        const int idx = t + 256 * i;
        const int r   = idx >> 7;
        const int c4  = idx & 127;
        const v4f v = *reinterpret_cast<const v4f*>(&lds[r * PITCH + c4 * 4]);
        __builtin_nontemporal_store(
            v, reinterpret_cast<v4f*>(dst + (size_t)r * COLS + c4 * 4));
    }
}

extern "C" void kernel_launch(void* const* d_in, const int* in_sizes, int n_in,
                              void* d_out, int out_size, void* d_ws, size_t ws_size,
                              hipStream_t stream) {
    (void)in_sizes; (void)n_in; (void)d_ws; (void)ws_size; (void)out_size;
    const float* x   = (const float*)d_in[0];
    float*       out = (float*)d_out;
    // 512 row-strips * 8 col-strips = 4096 workgroups of 256 threads (8 wave32).
    dim3 grid(4096), block(256);
    identity_mm_wmma<<<grid, block, 0, stream>>>(x, out);
}